// RegionProposalNetwork_77910706749626
// MI455X (gfx1250) — compile-verified
//
#include <hip/hip_runtime.h>
#include <stdint.h>
#include <math.h>

typedef float v2f __attribute__((ext_vector_type(2)));
typedef float v8f __attribute__((ext_vector_type(8)));
typedef unsigned int u32x4 __attribute__((ext_vector_type(4)));
typedef int i32x4 __attribute__((ext_vector_type(4)));
typedef int i32x8 __attribute__((ext_vector_type(8)));

#define HH 38
#define WW 38
#define NPIX (HH*WW)            // 1444
#define PT64 23                 // ceil(1444/64)
#define BS 8
#define CIN 512
#define COUT 512
#define KTOT (CIN*9)            // 4608
#define NANCH 9
#define NA (NPIX*NANCH)         // 12996
#define NSORT 16384
#define NPRE 12000
#define NPOST 2000
#define NEGF (-1e30f)

// ---- d_out layout (floats), reference return order ----
#define OUT_ROIS 0
#define OUT_RIDX (BS*NPOST*4)             // 64000
#define OUT_OFF  (OUT_RIDX + BS*NPOST)    // 80000
#define OUT_CLS  (OUT_OFF + BS*NA*4)      // 495872
#define OUT_ANC  (OUT_CLS + BS*NA*2)      // 703808

#if defined(__gfx1250__) && __has_builtin(__builtin_amdgcn_tensor_load_to_lds) && \
    __has_builtin(__builtin_amdgcn_s_wait_tensorcnt)
#define USE_TDM 1
#else
#define USE_TDM 0
#endif

// ---------------------------------------------------------------------------
// Anchors: [NA,4] (y1,x1,y2,x2), index = (y*38+x)*9 + a, a = r_idx*3 + s_idx
// ---------------------------------------------------------------------------
__global__ void k_anchors(float* __restrict__ anc) {
    int i = blockIdx.x * blockDim.x + threadIdx.x;
    if (i >= NA) return;
    int p = i / NANCH, a = i % NANCH;
    int y = p / WW, x = p % WW;
    int ri = a / 3, si = a % 3;
    float r = (ri == 0) ? 0.5f : (ri == 1 ? 1.0f : 2.0f);
    float s = (si == 0) ? 8.0f : (si == 1 ? 16.0f : 32.0f);
    float hb = 16.0f * s * sqrtf(r);
    float wb = 16.0f * s * sqrtf(1.0f / r);
    float sy = y * 16.0f, sx = x * 16.0f;
    anc[i*4+0] = sy + 8.0f - 0.5f*hb;
    anc[i*4+1] = sx + 8.0f - 0.5f*wb;
    anc[i*4+2] = sy + 8.0f + 0.5f*hb;
    anc[i*4+3] = sx + 8.0f + 0.5f*wb;
}

// ---------------------------------------------------------------------------
// conv1 3x3 + bias + ReLU via implicit GEMM on V_WMMA_F32_16X16X4_F32.
// Block = 128 thr (4 waves); block tile = 64 co x 64 pix.
// Wave w: co half = (w&1)*32, pix half = (w>>1)*32; 2x2 register tile
// (4 independent v8f accumulators -> 4 overlapping WMMA dep chains,
//  1 ds_load_b64 per WMMA).
// A tile (64co x 64k of W) staged by TDM tensor_load_to_lds when available
// (pad_enable reproduces the 66-float bank-padded row stride); B tile is the
// im2col gather. h stored [img][pix][512] so the 1x1 heads read coalesced.
// ---------------------------------------------------------------------------
__global__ __launch_bounds__(128)
void k_conv3x3(const float* __restrict__ X, const float* __restrict__ W,
               const float* __restrict__ Bias, float* __restrict__ H) {
    __shared__ float Als[64 * 66];       // [co_local][k], stride 66 (bank pad)
    __shared__ float Bls[32 * 64 * 2];   // [k/2][pix 0..63][2] b64-pair layout

    const int tid  = threadIdx.x;
    const int wv   = tid >> 5;
    const int lane = tid & 31;
    const int m    = lane & 15;
    const int half = lane >> 4;
    const int coh  = (wv & 1) * 32;
    const int pixh = (wv >> 1) * 32;

    const int bx     = blockIdx.x;         // 0 .. BS*PT64-1
    const int img    = bx / PT64;
    const int tile   = bx % PT64;
    const int pbase  = tile * 64;
    const int cobase = blockIdx.y * 64;

    v8f c00 = {0.f,0.f,0.f,0.f,0.f,0.f,0.f,0.f};
    v8f c01 = c00, c10 = c00, c11 = c00;

    // per-thread fixed im2col pixel for the B gather
    const int gpix = tid & 63;
    const int ghi  = tid >> 6;
    const int gp   = pbase + gpix;
    const bool gpv = gp < NPIX;
    const int gy = gp / WW, gx = gp % WW;

    for (int kbase = 0; kbase < KTOT; kbase += 64) {
        __syncthreads();
#if USE_TDM
        if (wv == 0) {
            unsigned long long ga =
                (unsigned long long)(const void*)(W + (size_t)cobase * KTOT + kbase);
            unsigned int lds_off = (unsigned int)(unsigned long long)(const void*)&Als[0];
            // D# group 0: count=1 | lds_addr | global_addr | type=2
            u32x4 g0 = { 1u, lds_off,
                         (unsigned)(ga & 0xFFFFFFFFu),
                         (unsigned)(((ga >> 32) & 0x01FFFFFFu) | 0x80000000u) };
            // D# group 1: data_size=4B, pad_enable, pad every 64 DW by 2 DW,
            // tensor_dim0=KTOT, tensor_dim1=COUT, tile 64x64, dim0_stride=KTOT
            i32x8 g1 = { (int)((2u << 16) | (1u << 20) | (5u << 22) | (1u << 25)),
                         (int)((unsigned)(KTOT & 0xFFFF) << 16),
                         (int)((unsigned)(COUT & 0xFFFF) << 16),
                         (int)(64u << 16),
                         64, KTOT, 0, 0 };
            i32x4 gz = {0, 0, 0, 0};
#if __clang_major__ >= 23
            i32x8 gz8 = {0, 0, 0, 0, 0, 0, 0, 0};
            __builtin_amdgcn_tensor_load_to_lds(g0, g1, gz, gz, gz8, 0);
#else
            __builtin_amdgcn_tensor_load_to_lds(g0, g1, gz, gz, 0);
#endif
        }
#else
        #pragma unroll 4
        for (int i = 0; i < 32; ++i) {
            int e = i * 128 + tid;
            int row = e >> 6, col = e & 63;
            Als[row * 66 + col] = W[(cobase + row) * KTOT + kbase + col];
        }
        if (kbase + 64 < KTOT)
            __builtin_prefetch(&W[(cobase + (tid >> 1)) * KTOT + kbase + 64], 0, 1);
#endif
        // Stage B: im2col gather, 64 k-values x 64 pixels (kk = 2*i + ghi)
        #pragma unroll 4
        for (int i = 0; i < 32; ++i) {
            int kk = i * 2 + ghi;
            int k = kbase + kk;
            int ci = k / 9, rem = k % 9;
            int ky = rem / 3, kx = rem % 3;
            float v = 0.0f;
            if (gpv) {
                int iy = gy + ky - 1, ix = gx + kx - 1;
                if (iy >= 0 && iy < HH && ix >= 0 && ix < WW)
                    v = X[((img * CIN + ci) * HH + iy) * WW + ix];
            }
            Bls[(i * 64 + gpix) * 2 + ghi] = v;
        }
#if USE_TDM
        if (wv == 0) __builtin_amdgcn_s_wait_tensorcnt(0);
#endif
        __syncthreads();

        // 16 WMMA steps of K=4, 2x2 register tile -> 4 WMMAs/step
        #pragma unroll
        for (int s = 0; s < 16; ++s) {
            int k0 = s * 4 + half * 2;
            v2f a0 = *(const v2f*)&Als[(coh      + m) * 66 + k0];
            v2f a1 = *(const v2f*)&Als[(coh + 16 + m) * 66 + k0];
            v2f b0 = *(const v2f*)&Bls[((k0 >> 1) * 64 + pixh      + m) * 2];
            v2f b1 = *(const v2f*)&Bls[((k0 >> 1) * 64 + pixh + 16 + m) * 2];
            c00 = __builtin_amdgcn_wmma_f32_16x16x4_f32(false, a0, false, b0, (short)0, c00, false, false);
            c01 = __builtin_amdgcn_wmma_f32_16x16x4_f32(false, a0, false, b1, (short)0, c01, false, false);
            c10 = __builtin_amdgcn_wmma_f32_16x16x4_f32(false, a1, false, b0, (short)0, c10, false, false);
            c11 = __builtin_amdgcn_wmma_f32_16x16x4_f32(false, a1, false, b1, (short)0, c11, false, false);
        }
    }

    // Epilogue. C/D layout: lanes 0-15: M=r, N=lane; lanes 16-31: M=r+8.
    // co indices r=0..7 are consecutive -> two b128 stores per accumulator.
    #pragma unroll
    for (int ch = 0; ch < 2; ++ch) {
        #pragma unroll
        for (int ph = 0; ph < 2; ++ph) {
            v8f c = (ch == 0) ? (ph == 0 ? c00 : c01) : (ph == 0 ? c10 : c11);
            int p = pbase + pixh + ph * 16 + m;
            if (p < NPIX) {
                int co0 = cobase + coh + ch * 16 + half * 8;
                float* o = &H[(size_t)(img * NPIX + p) * COUT + co0];
                float4 lo, hi;
                lo.x = fmaxf(c[0] + Bias[co0 + 0], 0.0f);
                lo.y = fmaxf(c[1] + Bias[co0 + 1], 0.0f);
                lo.z = fmaxf(c[2] + Bias[co0 + 2], 0.0f);
                lo.w = fmaxf(c[3] + Bias[co0 + 3], 0.0f);
                hi.x = fmaxf(c[4] + Bias[co0 + 4], 0.0f);
                hi.y = fmaxf(c[5] + Bias[co0 + 5], 0.0f);
                hi.z = fmaxf(c[6] + Bias[co0 + 6], 0.0f);
                hi.w = fmaxf(c[7] + Bias[co0 + 7], 0.0f);
                *(float4*)(o)     = lo;
                *(float4*)(o + 4) = hi;
            }
        }
    }
}

// ---------------------------------------------------------------------------
// 1x1 heads + softmax fg prob + bbox decode/clip/min-size + sort-key emit.
// One 64-thread block per (img,pixel); h column staged to LDS (coalesced).
// ---------------------------------------------------------------------------
__global__ __launch_bounds__(64)
void k_heads(const float* __restrict__ H,
             const float* __restrict__ Wc, const float* __restrict__ Bc,
             const float* __restrict__ Wb, const float* __restrict__ Bb,
             const float* __restrict__ anchors,
             const int* __restrict__ img_h, const int* __restrict__ img_w,
             const int* __restrict__ scale,
             float* __restrict__ out_off, float* __restrict__ out_cls,
             float* __restrict__ boxes, float* __restrict__ scores,
             unsigned long long* __restrict__ keys) {
    __shared__ float hs[CIN];
    __shared__ float outs[54];
    const int bid = blockIdx.x;
    const int img = bid / NPIX, p = bid % NPIX;
    const int t = threadIdx.x;

    const float* hp = H + (size_t)(img * NPIX + p) * CIN;
    #pragma unroll
    for (int i = 0; i < CIN / 64; ++i) hs[i * 64 + t] = hp[i * 64 + t];
    __syncthreads();

    if (t < 54) {
        const float* wrow;
        float b;
        if (t < 18) { wrow = Wc + t * CIN;        b = Bc[t]; }
        else        { wrow = Wb + (t - 18) * CIN; b = Bb[t - 18]; }
        float acc = 0.0f;
        for (int ci = 0; ci < CIN; ci += 4) {
            acc += wrow[ci]   * hs[ci]   + wrow[ci+1] * hs[ci+1]
                 + wrow[ci+2] * hs[ci+2] + wrow[ci+3] * hs[ci+3];
        }
        outs[t] = acc + b;
    }
    __syncthreads();

    if (t < NANCH) {
        const int a  = t;
        const int ai = p * NANCH + a;
        const int gi = img * NA + ai;
        float c0 = outs[a * 2], c1 = outs[a * 2 + 1];
        out_cls[gi * 2 + 0] = c0;
        out_cls[gi * 2 + 1] = c1;
        float mx = fmaxf(c0, c1);
        float e0 = expf(c0 - mx), e1 = expf(c1 - mx);
        float prob = e1 / (e0 + e1);

        float dy = outs[18 + a*4 + 0], dx = outs[18 + a*4 + 1];
        float dh = outs[18 + a*4 + 2], dw = outs[18 + a*4 + 3];
        out_off[gi*4+0] = dy; out_off[gi*4+1] = dx;
        out_off[gi*4+2] = dh; out_off[gi*4+3] = dw;

        const float* an = anchors + (size_t)ai * 4;
        float ha = an[2] - an[0], wa = an[3] - an[1];
        float cy = an[0] + 0.5f * ha, cx = an[1] + 0.5f * wa;
        float ncy = dy * ha + cy, ncx = dx * wa + cx;
        float nh = expf(dh) * ha, nw = expf(dw) * wa;
        float fh = (float)(*img_h), fw = (float)(*img_w);
        float y1 = fminf(fmaxf(ncy - 0.5f * nh, 0.0f), fh);
        float x1 = fminf(fmaxf(ncx - 0.5f * nw, 0.0f), fw);
        float y2 = fminf(fmaxf(ncy + 0.5f * nh, 0.0f), fh);
        float x2 = fminf(fmaxf(ncx + 0.5f * nw, 0.0f), fw);
        float msz = 16.0f * (float)(*scale);
        bool ok = ((y2 - y1) >= msz) && ((x2 - x1) >= msz);
        float s = ok ? prob : NEGF;

        float* bxp = boxes + (size_t)gi * 4;
        bxp[0] = y1; bxp[1] = x1; bxp[2] = y2; bxp[3] = x2;
        scores[gi] = s;

        unsigned u = __float_as_uint(s);
        unsigned ord = (u & 0x80000000u) ? ~u : (u | 0x80000000u);
        keys[((size_t)img << 14) + ai] =
            ((unsigned long long)(~ord) << 32) | (unsigned)ai;
    }
}

// ---------------------------------------------------------------------------
// Sort scaffolding
// ---------------------------------------------------------------------------
__global__ void k_init_keys(unsigned long long* __restrict__ keys) {
    int i = blockIdx.x * blockDim.x + threadIdx.x;
    if (i < BS * NSORT) keys[i] = 0xFFFFFFFFFFFFFFFFull;  // pad sorts last
}

__global__ void k_bitonic(unsigned long long* __restrict__ keys,
                          int size, int stride) {
    int t = blockIdx.x * blockDim.x + threadIdx.x;
    if (t >= BS * NSORT) return;
    int img = t >> 14;
    int i = t & (NSORT - 1);
    int j = i ^ stride;
    if (j <= i) return;
    unsigned long long* K = keys + ((size_t)img << 14);
    unsigned long long a = K[i], b = K[j];
    bool asc = (i & size) == 0;
    if (asc ? (a > b) : (a < b)) { K[i] = b; K[j] = a; }
}

// ---------------------------------------------------------------------------
// Greedy NMS: one block per image; kept boxes cached in LDS (40 KB).
// Equivalent to reference argmax-scan because candidates are score-sorted.
// ---------------------------------------------------------------------------
__global__ __launch_bounds__(256)
void k_nms(const unsigned long long* __restrict__ keys,
           const float* __restrict__ boxes, const float* __restrict__ scores,
           float* __restrict__ out_rois, float* __restrict__ out_idx) {
    __shared__ float kb[NPOST][4];
    __shared__ float karea[NPOST];
    __shared__ float cb[5];
    __shared__ int kept, sup, cand_ok;

    const int img = blockIdx.x;
    const int t = threadIdx.x;

    for (int i = t; i < NPOST; i += 256) {
        float* o = out_rois + (size_t)(img * NPOST + i) * 4;
        o[0] = 0.0f; o[1] = 0.0f; o[2] = 0.0f; o[3] = 0.0f;
        out_idx[img * NPOST + i] = -1.0f;
    }
    if (t == 0) kept = 0;
    __syncthreads();

    const unsigned long long* K = keys + ((size_t)img << 14);
    for (int j = 0; j < NPRE; ++j) {
        if (t == 0) {
            unsigned idx = (unsigned)(K[j] & 0xFFFFFFFFu);
            const float* b = boxes + ((size_t)img * NA + idx) * 4;
            float s = scores[(size_t)img * NA + idx];
            cb[0] = b[0]; cb[1] = b[1]; cb[2] = b[2]; cb[3] = b[3];
            cb[4] = (cb[2] - cb[0]) * (cb[3] - cb[1]);
            cand_ok = (s > 0.5f * NEGF) ? 1 : 0;
            sup = 0;
        }
        __syncthreads();
        if (!cand_ok) break;                    // uniform
        float y1 = cb[0], x1 = cb[1], y2 = cb[2], x2 = cb[3], ar = cb[4];
        int nk = kept;
        int local = 0;
        for (int k = t; k < nk; k += 256) {
            float yy1 = fmaxf(y1, kb[k][0]), xx1 = fmaxf(x1, kb[k][1]);
            float yy2 = fminf(y2, kb[k][2]), xx2 = fminf(x2, kb[k][3]);
            float inter = fmaxf(yy2 - yy1, 0.0f) * fmaxf(xx2 - xx1, 0.0f);
            float iou = inter / (ar + karea[k] - inter + 1e-9f);
            if (iou > 0.7f) local = 1;
        }
        if (local) sup = 1;                     // benign same-value race
        __syncthreads();
        int ssup = sup;
        if (!ssup && t == 0) {
            int k = kept;
            kb[k][0] = y1; kb[k][1] = x1; kb[k][2] = y2; kb[k][3] = x2;
            karea[k] = ar;
            float* o = out_rois + (size_t)(img * NPOST + k) * 4;
            o[0] = y1; o[1] = x1; o[2] = y2; o[3] = x2;
            out_idx[img * NPOST + k] = (float)img;
            kept = k + 1;
        }
        __syncthreads();
        if (kept >= NPOST) break;               // uniform
    }
}

// ---------------------------------------------------------------------------
extern "C" void kernel_launch(void* const* d_in, const int* in_sizes, int n_in,
                              void* d_out, int out_size, void* d_ws, size_t ws_size,
                              hipStream_t stream) {
    (void)in_sizes; (void)n_in; (void)out_size; (void)ws_size;
    const float* X  = (const float*)d_in[0];
    const float* W1 = (const float*)d_in[1];
    const float* B1 = (const float*)d_in[2];
    const float* Wc = (const float*)d_in[3];
    const float* Bc = (const float*)d_in[4];
    const float* Wb = (const float*)d_in[5];
    const float* Bb = (const float*)d_in[6];
    const int* img_h = (const int*)d_in[7];
    const int* img_w = (const int*)d_in[8];
    const int* scale = (const int*)d_in[9];
    float* out = (float*)d_out;

    // workspace layout (bytes)
    char* ws = (char*)d_ws;
    float* H       = (float*)ws;                            // 8*1444*512*4 = 23,658,496
    float* boxes   = (float*)(ws + 23658496);               // 8*12996*4*4  =  1,663,488
    float* scores  = (float*)(ws + 25321984);               // 8*12996*4    =    415,872
    unsigned long long* keys = (unsigned long long*)(ws + 25737856); // 8*16384*8 = 1,048,576

    float* out_rois = out + OUT_ROIS;
    float* out_ridx = out + OUT_RIDX;
    float* out_off  = out + OUT_OFF;
    float* out_cls  = out + OUT_CLS;
    float* out_anc  = out + OUT_ANC;

    k_anchors<<<(NA + 255) / 256, 256, 0, stream>>>(out_anc);
    k_conv3x3<<<dim3(BS * PT64, COUT / 64), 128, 0, stream>>>(X, W1, B1, H);
    k_init_keys<<<(BS * NSORT) / 256, 256, 0, stream>>>(keys);
    k_heads<<<BS * NPIX, 64, 0, stream>>>(H, Wc, Bc, Wb, Bb, out_anc,
                                          img_h, img_w, scale,
                                          out_off, out_cls, boxes, scores, keys);
    for (int size = 2; size <= NSORT; size <<= 1)
        for (int stride = size >> 1; stride > 0; stride >>= 1)
            k_bitonic<<<(BS * NSORT) / 256, 256, 0, stream>>>(keys, size, stride);
    k_nms<<<BS, 256, 0, stream>>>(keys, boxes, scores, out_rois, out_ridx);
}